// AnalyticalStage2_90804198572170
// MI455X (gfx1250) — compile-verified
//
#include <hip/hip_runtime.h>
#include <math.h>
#include <stdint.h>

typedef __attribute__((ext_vector_type(2))) float v2f;
typedef __attribute__((ext_vector_type(8))) float v8f;
typedef __attribute__((ext_vector_type(4))) int   v4i;

typedef __attribute__((address_space(1))) v4i* gas_v4i_ptr;
typedef __attribute__((address_space(3))) v4i* las_v4i_ptr;

#define NT       32768
#define THREADS  256
#define TILE     2048          // elements per workgroup tile (256 thr * 8)
#define NTILES   (NT / TILE)   // 16
#define CHUNKS   128           // 16-element chunks per tile
#define CONST_C  0.206756f

// ---- CDNA5 async global->LDS copy (ASYNCcnt path), 16B per lane ----
static __device__ __forceinline__ void async_ld_b128(const float* g, float* l) {
#if __has_builtin(__builtin_amdgcn_global_load_async_to_lds_b128)
    __builtin_amdgcn_global_load_async_to_lds_b128(
        (gas_v4i_ptr)g, (las_v4i_ptr)l, 0, 0);
#else
    uint32_t loff = (uint32_t)(uintptr_t)l;     // generic LDS ptr: low 32 = LDS offset
    uint64_t ga   = (uint64_t)(uintptr_t)g;
    asm volatile("global_load_async_to_lds_b128 %0, %1, off"
                 :: "v"(loff), "v"(ga) : "memory");
#endif
}

static __device__ __forceinline__ void wait_async0() {
#if __has_builtin(__builtin_amdgcn_s_wait_asynccnt)
    __builtin_amdgcn_s_wait_asynccnt(0);
#else
    asm volatile("s_wait_asynccnt 0" ::: "memory");
#endif
}

// One workgroup per batch row. Scan-as-matmul:
//   S_loc = L x DP (L[m][k] = d^(m-k), lower triangular) via 4 chained
//   v_wmma_f32_16x16x4_f32 per wave; async-double-buffered p tiles.
__global__ __launch_bounds__(THREADS)
void visco_scan_wmma(const float* __restrict__ t,
                     const float* __restrict__ p,
                     const float* __restrict__ h_raw,
                     float* __restrict__ out)
{
    __shared__ __align__(16) float bufp[2][TILE];  // double-buffered p tiles
    __shared__ __align__(16) float lds_o[TILE];
    __shared__ float totals[CHUNKS];
    __shared__ float enter[CHUNKS];
    __shared__ float grp[8];
    __shared__ float gent[8];
    __shared__ float lds_e;                        // cross-tile scan carry

    const int b    = blockIdx.x;
    const int tid  = threadIdx.x;
    const int lane = tid & 31;
    const int w    = tid >> 5;
    const int col  = lane & 15;                    // chunk (N) index in wave tile
    const int hh   = lane >> 4;                    // lane half

    const float E1  = h_raw[b * 3 + 0];
    const float E2  = h_raw[b * 3 + 1];
    const float eta = h_raw[b * 3 + 2];
    const float alpha = E1 * E2 / ((E1 + E2) * eta);
    const float Dc    = CONST_C * E1 / (E2 * (E1 + E2));
    const float AD    = CONST_C / E2;              // A + D
    const float dtv   = t[1] - t[0];               // linspace step
    const float aa    = alpha * dtv;               // per-step decay exponent

    // L rows: L[m][k] = exp(-aa*(m-k)) for k<=m else 0, m = col
    float Lrow[16];
#pragma unroll
    for (int k = 0; k < 16; ++k) {
        int dlt = col - k;
        Lrow[k] = (dlt >= 0) ? __expf(-aa * (float)dlt) : 0.0f;
    }
    // A operand per K-step: VGPR0 = K {0|2}+4k0, VGPR1 = K {1|3}+4k0
    v2f Aop[4];
#pragma unroll
    for (int k0 = 0; k0 < 4; ++k0) {
        Aop[k0].x = hh ? Lrow[4 * k0 + 2] : Lrow[4 * k0 + 0];
        Aop[k0].y = hh ? Lrow[4 * k0 + 3] : Lrow[4 * k0 + 1];
    }
    const float e8    = __expf(-aa * 8.0f);
    const float baseh = hh ? e8 : 1.0f;
    float dpw[8];                                  // d^(r+1)
#pragma unroll
    for (int r = 0; r < 8; ++r) dpw[r] = __expf(-aa * (float)(r + 1));
    const float d16  = __expf(-aa * 16.0f);
    const float d256 = __expf(-aa * 256.0f);

    if (tid == 0) lds_e = 0.0f;

    const float* prow = p   + (size_t)b * NT;
    float*       orow = out + (size_t)b * NT;
    const int woff = w * 256;                      // wave sub-tile offset

    // prefetch tile 0 into bufp[0]
#pragma unroll
    for (int i = 0; i < 2; ++i)
        async_ld_b128(prow + 4 * (tid + THREADS * i),
                      &bufp[0][4 * (tid + THREADS * i)]);
    float prevp = 0.0f;                            // p[-1] treated as 0

    for (int tile = 0; tile < NTILES; ++tile) {
        const int cur = tile & 1, nxt = cur ^ 1;
        const int tbase = tile * TILE;

        wait_async0();
        __syncthreads();                           // bufp[cur] valid everywhere

        const float nextprev = bufp[cur][TILE - 1];
        if (tile + 1 < NTILES) {                   // overlap: prefetch next tile
#pragma unroll
            for (int i = 0; i < 2; ++i)
                async_ld_b128(prow + tbase + TILE + 4 * (tid + THREADS * i),
                              &bufp[nxt][4 * (tid + THREADS * i)]);
        }

        // local scans: S_loc = L x DP, 16x16x16 via 4 chained 16x16x4 WMMAs
        v8f c = {0.f, 0.f, 0.f, 0.f, 0.f, 0.f, 0.f, 0.f};
#pragma unroll
        for (int k0 = 0; k0 < 4; ++k0) {
            const int K0 = 4 * k0 + 2 * hh;
            const int i0 = woff + 16 * col + K0;
            const float pm1 = (i0 > 0) ? bufp[cur][i0 - 1] : prevp;
            v2f bv;
            bv.x = bufp[cur][i0]     - pm1;            // dp at K0
            bv.y = bufp[cur][i0 + 1] - bufp[cur][i0];  // dp at K0+1
            c = __builtin_amdgcn_wmma_f32_16x16x4_f32(
                    false, Aop[k0], false, bv, (short)0, c, false, false);
        }

        // chunk totals = row m=15 of S_loc: lanes 16..31, VGPR 7
        if (lane >= 16) totals[w * 16 + (lane - 16)] = c[7];
        __syncthreads();

        // two-level carry scan over 128 chunks (constant multipliers)
        if (tid < 8) {                             // group totals (d^16 chain)
            float g = 0.0f;
#pragma unroll
            for (int j = 0; j < 16; ++j) g = __fmaf_rn(d16, g, totals[tid * 16 + j]);
            grp[tid] = g;
        }
        __syncthreads();
        if (tid == 0) {                            // scan 8 groups (d^256 chain)
            float e = lds_e;
#pragma unroll
            for (int gi = 0; gi < 8; ++gi) { gent[gi] = e; e = __fmaf_rn(d256, e, grp[gi]); }
            lds_e = e;                             // carry into next tile
        }
        __syncthreads();
        if (tid < 8) {                             // expand to per-chunk enters
            float e = gent[tid];
#pragma unroll
            for (int j = 0; j < 16; ++j) {
                enter[tid * 16 + j] = e;
                e = __fmaf_rn(d16, e, totals[tid * 16 + j]);
            }
        }
        __syncthreads();

        // finalize: s[m] = S_loc[m] + d^(m+1)*enter ; omega = AD*p - D*s
        const float ent = enter[w * 16 + col];
#pragma unroll
        for (int r = 0; r < 8; ++r) {
            const int idx = woff + 16 * col + r + 8 * hh;   // m = r + 8*hh
            const float s = __fmaf_rn(dpw[r] * baseh, ent, c[r]);
            lds_o[idx] = __fmaf_rn(AD, bufp[cur][idx], -Dc * s);
        }
        __syncthreads();

        // coalesced 16B omega stores
#pragma unroll
        for (int i = 0; i < 2; ++i) {
            const float4 v = *(const float4*)&lds_o[4 * (tid + THREADS * i)];
            *(float4*)&orow[tbase + 4 * (tid + THREADS * i)] = v;
        }
        prevp = nextprev;
    }
}

extern "C" void kernel_launch(void* const* d_in, const int* in_sizes, int n_in,
                              void* d_out, int out_size, void* d_ws, size_t ws_size,
                              hipStream_t stream) {
    // setup_inputs order: h (unused), t, p, h_raw
    const float* t     = (const float*)d_in[1];
    const float* p     = (const float*)d_in[2];
    const float* h_raw = (const float*)d_in[3];
    float*       out   = (float*)d_out;
    const int B = in_sizes[2] / NT;   // 512
    hipLaunchKernelGGL(visco_scan_wmma, dim3(B), dim3(THREADS), 0, stream,
                       t, p, h_raw, out);
}